// NormalDistributionChecker1D_49787260895473
// MI455X (gfx1250) — compile-verified
//
#include <hip/hip_runtime.h>

typedef __attribute__((ext_vector_type(2))) float v2f;
typedef __attribute__((ext_vector_type(8))) float v8f;

#define NBLK 1024
#define NTHR 256
#define NWAVE (NTHR / 32)

__constant__ float g_zscores[9] = {-1.2815516f, -0.8416212f, -0.5244005f, -0.2533471f, 0.0f,
                                   0.2533471f, 0.5244005f, 0.8416212f, 1.2815516f};
__constant__ float g_crit[9] = {14.683657f, 12.242145f, 10.656372f, 9.413640f, 8.342832f,
                                7.357034f, 6.393306f, 5.380053f, 4.168159f};
__constant__ float g_quant[9] = {0.1f, 0.2f, 0.3f, 0.4f, 0.5f, 0.6f, 0.7f, 0.8f, 0.9f};

__device__ __forceinline__ float wave_sum(float v) {
#pragma unroll
  for (int off = 16; off > 0; off >>= 1) v += __shfl_down(v, (unsigned)off, 32);
  return v;  // valid in lane 0
}

// Reduce NBLK consecutive floats with V_WMMA_F32_16X16X4_F32 (A = ones).
// Every row of D accumulates the per-column sums of B; lane-to-(K,N) mapping is
// irrelevant for a total sum. VGPR0 of D holds rows 0 and 8 -> total = sum/2.
__device__ __forceinline__ float wmma_reduce_1024(const float* __restrict__ src, int lane) {
  v2f ones;
  ones[0] = 1.0f;
  ones[1] = 1.0f;
  v8f c = {};
#pragma unroll 4
  for (int t = 0; t < NBLK; t += 64) {
    v2f b;
    b[0] = src[t + lane];
    b[1] = src[t + 32 + lane];
    c = __builtin_amdgcn_wmma_f32_16x16x4_f32(false, ones, false, b, (short)0, c, false, false);
  }
  return 0.5f * wave_sum(c[0]);  // valid in lane 0
}

// ---------------- Pass 1: per-block sum and sum-of-squares ----------------
__global__ void ndc_pass1(const float* __restrict__ x, int n, float* __restrict__ part) {
  const int tid = blockIdx.x * blockDim.x + threadIdx.x;
  const int stride = gridDim.x * blockDim.x;
  const int n4 = n >> 2;
  const float4* __restrict__ x4 = (const float4*)x;
  float s = 0.0f, s2 = 0.0f;
  for (int i = tid; i < n4; i += stride) {
    __builtin_prefetch(&x4[i + stride], 0, 0);  // global_prefetch_b8
    float4 v = x4[i];
    s += (v.x + v.y) + (v.z + v.w);
    s2 = fmaf(v.x, v.x, s2);
    s2 = fmaf(v.y, v.y, s2);
    s2 = fmaf(v.z, v.z, s2);
    s2 = fmaf(v.w, v.w, s2);
  }
  for (int i = (n4 << 2) + tid; i < n; i += stride) {
    float v = x[i];
    s += v;
    s2 = fmaf(v, v, s2);
  }
  __shared__ float ls[NWAVE], ls2[NWAVE];
  s = wave_sum(s);
  s2 = wave_sum(s2);
  const int lane = threadIdx.x & 31, wid = threadIdx.x >> 5;
  if (lane == 0) {
    ls[wid] = s;
    ls2[wid] = s2;
  }
  __syncthreads();
  if (threadIdx.x == 0) {
    float a = 0.0f, b = 0.0f;
#pragma unroll
    for (int w = 0; w < NWAVE; ++w) {
      a += ls[w];
      b += ls2[w];
    }
    part[blockIdx.x] = a;
    part[NBLK + blockIdx.x] = b;
  }
}

// ------------- Stats: WMMA-reduce partials, fold affine transform -------------
__global__ void ndc_stats(const float* __restrict__ part, float n, float* __restrict__ stats) {
  const int lane = threadIdx.x;
  float sum = wmma_reduce_1024(part, lane);
  float sumsq = wmma_reduce_1024(part + NBLK, lane);
  if (lane == 0) {
    float mean = sum / n;
    float var = (sumsq - sum * sum / n) / (n - 1.0f);  // ddof=1
    float rstd = rsqrtf(var);
    float scale = 100.0f * rstd;  // sharpness / std
    stats[0] = scale;
#pragma unroll
    for (int k = 0; k < 9; ++k) stats[1 + k] = fmaf(scale, mean, 100.0f * g_zscores[k]);
  }
}

// --------- Pass 2: 9 soft-histogram sums; sigmoid(arg)=rcp(1+exp(-arg)) ---------
__global__ void ndc_pass2(const float* __restrict__ x, int n, const float* __restrict__ stats,
                          float* __restrict__ part) {
  const float scale = stats[0];
  float ck[9];
#pragma unroll
  for (int k = 0; k < 9; ++k) ck[k] = stats[1 + k];
  float acc[9];
#pragma unroll
  for (int k = 0; k < 9; ++k) acc[k] = 0.0f;

  const int tid = blockIdx.x * blockDim.x + threadIdx.x;
  const int stride = gridDim.x * blockDim.x;
  const int n4 = n >> 2;
  const float4* __restrict__ x4 = (const float4*)x;
  for (int i = tid; i < n4; i += stride) {
    __builtin_prefetch(&x4[i + stride], 0, 0);
    float4 v = x4[i];
    float tv[4] = {v.x * scale, v.y * scale, v.z * scale, v.w * scale};
#pragma unroll
    for (int e = 0; e < 4; ++e) {
#pragma unroll
      for (int k = 0; k < 9; ++k) {
        // arg = ck - scale*x ; sigmoid(arg) = 1/(1 + exp(scale*x - ck))
        float ex = __expf(tv[e] - ck[k]);
        acc[k] += __builtin_amdgcn_rcpf(1.0f + ex);
      }
    }
  }
  for (int i = (n4 << 2) + tid; i < n; i += stride) {
    float t = x[i] * scale;
#pragma unroll
    for (int k = 0; k < 9; ++k) acc[k] += __builtin_amdgcn_rcpf(1.0f + __expf(t - ck[k]));
  }

  __shared__ float red[9][NWAVE];
  const int lane = threadIdx.x & 31, wid = threadIdx.x >> 5;
#pragma unroll
  for (int k = 0; k < 9; ++k) {
    float s = wave_sum(acc[k]);
    if (lane == 0) red[k][wid] = s;
  }
  __syncthreads();
  if (threadIdx.x < 9) {
    float s = 0.0f;
#pragma unroll
    for (int w = 0; w < NWAVE; ++w) s += red[threadIdx.x][w];
    part[threadIdx.x * NBLK + blockIdx.x] = s;
  }
}

// --------------- Final: WMMA-reduce 9 partial arrays, chi2 epilogue ---------------
__global__ void ndc_final(const float* __restrict__ part, float n, float* __restrict__ out) {
  const int lane = threadIdx.x;
  float cum[9];  // only lane 0's copy is meaningful
  for (int k = 0; k < 9; ++k) cum[k] = wmma_reduce_1024(part + k * NBLK, lane);

  if (lane == 0) {
    const float expected = 0.1f * n;
    const float inv_e = 1.0f / (expected + 1e-7f);
    float chi2 = 0.0f;
    float a = cum[0] - expected;
    chi2 += a * a * inv_e;
#pragma unroll
    for (int k = 1; k < 9; ++k) {
      a = (cum[k] - cum[k - 1]) - expected;
      chi2 += a * a * inv_e;
    }
    a = (n - cum[8]) - expected;
    chi2 += a * a * inv_e;

    // softmax(-|chi2 - CRIT|), max-subtracted for stability
    float xs[9], m = -1e30f;
#pragma unroll
    for (int k = 0; k < 9; ++k) {
      xs[k] = -fabsf(chi2 - g_crit[k]);
      m = fmaxf(m, xs[k]);
    }
    float se = 0.0f, sw = 0.0f;
#pragma unroll
    for (int k = 0; k < 9; ++k) {
      float e = __expf(xs[k] - m);
      se += e;
      sw += e * g_quant[k];
    }
    float p = 1.0f - sw / se;
    float excess = fmaxf(0.0f, (chi2 - 14.683657f) * 0.01f);
    out[0] = p + excess;
  }
}

extern "C" void kernel_launch(void* const* d_in, const int* in_sizes, int n_in,
                              void* d_out, int out_size, void* d_ws, size_t ws_size,
                              hipStream_t stream) {
  const float* x = (const float*)d_in[0];
  const int n = in_sizes[0];
  float* ws = (float*)d_ws;
  float* p1 = ws;                   // 2 * NBLK floats (sum, sumsq partials)
  float* stats = ws + 2 * NBLK;     // 16 floats (scale, c_k[9])
  float* p2 = ws + 2 * NBLK + 16;   // 9 * NBLK floats (per-threshold partials)

  ndc_pass1<<<NBLK, NTHR, 0, stream>>>(x, n, p1);
  ndc_stats<<<1, 32, 0, stream>>>(p1, (float)n, stats);
  ndc_pass2<<<NBLK, NTHR, 0, stream>>>(x, n, stats, p2);
  ndc_final<<<1, 32, 0, stream>>>(p2, (float)n, (float*)d_out);
}